// PolyConv_15814069584343
// MI455X (gfx1250) — compile-verified
//
#include <hip/hip_runtime.h>

// PolyConv (Chebyshev-like): h = theta0*f; for k=1..4: f = f - A f; h += theta_k f
// A: 1.6M COO edges over 100K nodes, 32 features (== wave32 width).
// Strategy: build CSR once (edge_row is invariant across the 4 iterations),
// then 4 gather-phase fused SpMM passes. feat (12.8MB) stays L2-resident.
// gfx1250 path: edge (col,val) stream double-buffered into LDS via
// GLOBAL_LOAD_ASYNC_TO_LDS_B64 (ASYNCcnt), consumed with LDS broadcasts.

#define NN 100000
#define NE 1600000
#define DF 32

#if defined(__HIP_DEVICE_COMPILE__) && defined(__gfx1250__) && \
    __has_builtin(__builtin_amdgcn_global_load_async_to_lds_b64) && \
    __has_builtin(__builtin_amdgcn_s_wait_asynccnt)
#define USE_ASYNC_LDS 1
#else
#define USE_ASYNC_LDS 0
#endif

#if USE_ASYNC_LDS
typedef int pc_v2i __attribute__((__vector_size__(8)));
typedef __attribute__((address_space(1))) pc_v2i* pc_gp;   // global (flat-identical)
typedef __attribute__((address_space(3))) pc_v2i* pc_lp;   // LDS (32-bit offset)

__device__ __forceinline__ void pc_async_pair(const uint2* g, uint2* l) {
    // AS1 pointer: same bit pattern as the flat pointer.
    // AS3 pointer: low 32 bits of the flat shared pointer.
    __builtin_amdgcn_global_load_async_to_lds_b64(
        (pc_gp)(uintptr_t)g,
        (pc_lp)(unsigned)(uintptr_t)l,
        /*offset=*/0, /*cpol=*/0);
}
#endif

// ---------------- hot kernel first (so the disasm snippet shows it) ----------------
// One wave per node, lane == feature (DF == 32 == wave32 width).
// fout = fin - A*fin  (gather over the node's contiguous CSR segment)
// h   += theta[k] * fout
__global__ void pc_spmm(const uint2* __restrict__ ep, const int* __restrict__ row_ptr,
                        const float* __restrict__ fin, float* __restrict__ fout,
                        float* __restrict__ h, const float* __restrict__ theta, int k) {
    __shared__ uint2 stage[8][2][32];   // per-wave double buffer, 4KB/block

    int lane = threadIdx.x & 31;
    int wid  = threadIdx.x >> 5;
    int node = blockIdx.x * 8 + wid;
    if (node >= NN) return;

    int s = row_ptr[node];
    int e = row_ptr[node + 1];
    int deg = e - s;
    float th = theta[k];
    float acc = 0.0f;

#if USE_ASYNC_LDS
    if (deg > 0) {
        int nchunks = (deg + 31) >> 5;
        // prologue: stage chunk 0 (clamp tail lanes to a valid edge)
        {
            int gi = s + lane; if (gi >= e) gi = e - 1;
            pc_async_pair(ep + gi, &stage[wid][0][lane]);
        }
        for (int c = 0; c < nchunks; ++c) {
            if (c + 1 < nchunks) {
                int gi = s + (c + 1) * 32 + lane; if (gi >= e) gi = e - 1;
                pc_async_pair(ep + gi, &stage[wid][(c + 1) & 1][lane]);
                __builtin_amdgcn_s_wait_asynccnt(1);   // chunk c has landed
            } else {
                __builtin_amdgcn_s_wait_asynccnt(0);
            }
            asm volatile("" ::: "memory");
            int cnt = deg - c * 32; if (cnt > 32) cnt = 32;
            const uint2* buf = stage[wid][c & 1];
            int j = 0;
            for (; j + 4 <= cnt; j += 4) {              // LDS broadcast reads
                uint2 p0 = buf[j + 0];
                uint2 p1 = buf[j + 1];
                uint2 p2 = buf[j + 2];
                uint2 p3 = buf[j + 3];
                acc += __uint_as_float(p0.y) * fin[(int)p0.x * DF + lane];
                acc += __uint_as_float(p1.y) * fin[(int)p1.x * DF + lane];
                acc += __uint_as_float(p2.y) * fin[(int)p2.x * DF + lane];
                acc += __uint_as_float(p3.y) * fin[(int)p3.x * DF + lane];
            }
            for (; j < cnt; ++j) {
                uint2 pr = buf[j];
                acc += __uint_as_float(pr.y) * fin[(int)pr.x * DF + lane];
            }
        }
    }
#else
    {
        int i = s;
        int e4 = s + ((e - s) & ~3);
        for (; i < e4; i += 4) {
            __builtin_prefetch(ep + i + 32, 0, 0);
            uint2 p0 = ep[i + 0];
            uint2 p1 = ep[i + 1];
            uint2 p2 = ep[i + 2];
            uint2 p3 = ep[i + 3];
            acc += __uint_as_float(p0.y) * fin[(int)p0.x * DF + lane];
            acc += __uint_as_float(p1.y) * fin[(int)p1.x * DF + lane];
            acc += __uint_as_float(p2.y) * fin[(int)p2.x * DF + lane];
            acc += __uint_as_float(p3.y) * fin[(int)p3.x * DF + lane];
        }
        for (; i < e; ++i) {
            uint2 pr = ep[i];
            acc += __uint_as_float(pr.y) * fin[(int)pr.x * DF + lane];
        }
    }
#endif

    int idx = node * DF + lane;
    float nf = fin[idx] - acc;
    fout[idx] = nf;
    h[idx] += th * nf;
}

// ---------------- CSR build ----------------

__global__ void pc_hist(const int* __restrict__ erow, int* __restrict__ counts) {
    int e = blockIdx.x * 256 + threadIdx.x;
    if (e < NE) atomicAdd(&counts[erow[e]], 1);
}

// per-256-block exclusive scan; emit block totals
__global__ void pc_scan1(const int* __restrict__ counts, int* __restrict__ row_ptr,
                         int* __restrict__ bsum) {
    __shared__ int sh[256];
    int i = blockIdx.x * 256 + threadIdx.x;
    int v = (i < NN) ? counts[i] : 0;
    sh[threadIdx.x] = v;
    __syncthreads();
    for (int off = 1; off < 256; off <<= 1) {
        int t = (threadIdx.x >= off) ? sh[threadIdx.x - off] : 0;
        __syncthreads();
        sh[threadIdx.x] += t;
        __syncthreads();
    }
    int incl = sh[threadIdx.x];
    if (i < NN) row_ptr[i] = incl - v;           // exclusive within block
    if (threadIdx.x == 255) bsum[blockIdx.x] = incl;
}

// single-block exclusive scan of block sums (nb <= 512)
__global__ void pc_scan2(int* __restrict__ bsum, int nb) {
    __shared__ int sh[512];
    int v = (threadIdx.x < nb) ? bsum[threadIdx.x] : 0;
    sh[threadIdx.x] = v;
    __syncthreads();
    for (int off = 1; off < 512; off <<= 1) {
        int t = (threadIdx.x >= off) ? sh[threadIdx.x - off] : 0;
        __syncthreads();
        sh[threadIdx.x] += t;
        __syncthreads();
    }
    if (threadIdx.x < nb) bsum[threadIdx.x] = sh[threadIdx.x] - v;  // exclusive
}

__global__ void pc_scan3(int* __restrict__ row_ptr, int* __restrict__ cursor,
                         const int* __restrict__ bsum) {
    int i = blockIdx.x * 256 + threadIdx.x;
    if (i < NN) {
        int v = row_ptr[i] + bsum[blockIdx.x];
        row_ptr[i] = v;
        cursor[i]  = v;
    }
    if (i == 0) row_ptr[NN] = NE;   // total is statically known
}

// bin edges into CSR order, packing (col, val) into one uint2 for 8B streams
__global__ void pc_scatter(const int* __restrict__ erow, const int* __restrict__ ecol,
                           const float* __restrict__ eval, int* __restrict__ cursor,
                           uint2* __restrict__ ep) {
    int e = blockIdx.x * 256 + threadIdx.x;
    if (e < NE) {
        int r = erow[e];
        int p = atomicAdd(&cursor[r], 1);
        ep[p] = make_uint2((unsigned)ecol[e], __float_as_uint(eval[e]));
    }
}

__global__ void pc_init_h(const float* __restrict__ feat, const float* __restrict__ theta,
                          float* __restrict__ h) {
    int i = blockIdx.x * 256 + threadIdx.x;
    if (i < NN * DF) h[i] = theta[0] * feat[i];
}

// ---------------- launch ----------------

static inline size_t al256(size_t x) { return (x + 255) & ~(size_t)255; }

extern "C" void kernel_launch(void* const* d_in, const int* in_sizes, int n_in,
                              void* d_out, int out_size, void* d_ws, size_t ws_size,
                              hipStream_t stream) {
    (void)in_sizes; (void)n_in; (void)out_size; (void)ws_size;

    const float* feat  = (const float*)d_in[0];  // [NN*DF]
    const float* evals = (const float*)d_in[1];  // [NE]
    const float* theta = (const float*)d_in[2];  // [5]
    const int*   erow  = (const int*)d_in[3];    // [NE]
    const int*   ecol  = (const int*)d_in[4];    // [NE]
    float*       h     = (float*)d_out;          // [NN*DF]

    // workspace carve-up
    char* p = (char*)d_ws;
    int*   counts  = (int*)p;              p += al256((size_t)NN * 4);
    int*   row_ptr = (int*)p;              p += al256((size_t)(NN + 1) * 4);
    int*   cursor  = (int*)p;              p += al256((size_t)NN * 4);
    int*   bsum    = (int*)p;              p += al256((size_t)512 * 4);
    uint2* ep      = (uint2*)p;            p += al256((size_t)NE * 8);
    float* feat_a  = (float*)p;            p += al256((size_t)NN * DF * 4);
    float* feat_b  = (float*)p;            p += al256((size_t)NN * DF * 4);

    const int nbN   = (NN + 255) / 256;        // 391 scan blocks (<=512)
    const int nbE   = (NE + 255) / 256;        // 6250
    const int nbF   = (NN * DF + 255) / 256;   // 12500
    const int nbSp  = (NN + 7) / 8;            // 12500 (8 nodes / 256-thread block)

    // ---- CSR build (once; edge_row invariant across all 4 iterations) ----
    (void)hipMemsetAsync(counts, 0, (size_t)NN * 4, stream);
    pc_hist   <<<nbE, 256, 0, stream>>>(erow, counts);
    pc_scan1  <<<nbN, 256, 0, stream>>>(counts, row_ptr, bsum);
    pc_scan2  <<<1,   512, 0, stream>>>(bsum, nbN);
    pc_scan3  <<<nbN, 256, 0, stream>>>(row_ptr, cursor, bsum);
    pc_scatter<<<nbE, 256, 0, stream>>>(erow, ecol, evals, cursor, ep);

    // ---- h = theta0 * feat ----
    pc_init_h <<<nbF, 256, 0, stream>>>(feat, theta, h);

    // ---- 4 fused (f = f - A f ; h += theta_k f) passes, ping-pong buffers ----
    pc_spmm<<<nbSp, 256, 0, stream>>>(ep, row_ptr, feat,   feat_b, h, theta, 1);
    pc_spmm<<<nbSp, 256, 0, stream>>>(ep, row_ptr, feat_b, feat_a, h, theta, 2);
    pc_spmm<<<nbSp, 256, 0, stream>>>(ep, row_ptr, feat_a, feat_b, h, theta, 3);
    pc_spmm<<<nbSp, 256, 0, stream>>>(ep, row_ptr, feat_b, feat_a, h, theta, 4);
}